// Conv_RBS_density_1632087572678
// MI455X (gfx1250) — compile-verified
//
#include <hip/hip_runtime.h>

// ---------------------------------------------------------------------------
// QCNN RBS density conjugation, restructured for MI455X (gfx1250, wave32):
//   rho_out = W rho W^T,   W = U_48 ... U_1  (each U a sparse Givens operator)
// Phase 0: zero-pad rho to 512x512 (rho is exactly symmetric -> transposed
//          access is free, giving contiguous b64 B-fragment loads).
// Phase 1: build W (496x496 in 512x512) via 48 sequential sparse Givens
//          row-pair rotations (single workgroup, L2-resident).
// Phase 2: T = W * rho_pad ; out = T * W^T with V_WMMA_F32_16X16X4_F32.
//          Both matmuls use the same transposed-B kernel (B(k,n) = Bp[n][k]).
// ---------------------------------------------------------------------------

typedef float v2f __attribute__((ext_vector_type(2)));
typedef float v8f __attribute__((ext_vector_type(8)));

#define NDIM 496   // C(32,2)
#define PDIM 512   // padded to 32 * 16 for exact WMMA tiling
#define NQ   32    // qubits

__device__ __forceinline__ int state_idx(int a, int b) {
  // lexicographic index of pair (a,b), a<b, over NQ qubits
  return a * (2 * NQ - a - 1) / 2 + (b - a - 1);
}

// ---------------- Phase 0: zero-pad rho into 512x512 -----------------------
__global__ __launch_bounds__(256)
void pad_rho_kernel(const float* __restrict__ rho, float* __restrict__ rp) {
  const int e = blockIdx.x * blockDim.x + threadIdx.x;  // 0 .. 512*512-1
  const int r = e >> 9;          // /512
  const int c = e & (PDIM - 1);  // %512
  rp[e] = (r < NDIM && c < NDIM) ? rho[r * NDIM + c] : 0.0f;
}

// ---------------- Phase 1: build W = product of Givens gates ---------------
__global__ __launch_bounds__(1024)
void build_w_kernel(const float* __restrict__ thetas,
                    const int* __restrict__ u_idx,
                    const int* __restrict__ p_idx,
                    int n_gates,
                    float* __restrict__ W) {
  const int tid = threadIdx.x;
  // W = zero-padded identity (512x512)
  for (int e = tid; e < PDIM * PDIM; e += blockDim.x) W[e] = 0.0f;
  __syncthreads();
  for (int d = tid; d < NDIM; d += blockDim.x) W[d * PDIM + d] = 1.0f;
  __syncthreads();

  for (int g = 0; g < n_gates; ++g) {
    const int u = u_idx[g];
    const int p = p_idx[g];
    // unique gate tuples are (q, q+1), q = 4*(u/3) + u%3 (sorted set)
    const int i = 4 * (u / 3) + (u % 3);
    const int j = i + 1;
    const float ct = cosf(thetas[p]);
    const float st = sinf(thetas[p]);
    // U row-mix: row(sA) <- ct*row(sA) + st*row(sB);  row(sB) <- -st*.. + ct*..
    // 30 disjoint state pairs (spectator qubits) x 496 live columns
    const int work = (NQ - 2) * NDIM;
    for (int w = tid; w < work; w += blockDim.x) {
      const int spec = w / NDIM;
      const int col  = w - spec * NDIM;
      const int c = (spec < i) ? spec : spec + 2;        // skip qubits i, i+1
      const int ia = c < i ? c : i, ib = c < i ? i : c;  // sorted (i, c)
      const int ja = c < j ? c : j, jb = c < j ? j : c;  // sorted (j, c)
      const int sA = state_idx(ia, ib);                  // state containing i
      const int sB = state_idx(ja, jb);                  // state containing j
      const float a = W[sA * PDIM + col];
      const float b = W[sB * PDIM + col];
      W[sA * PDIM + col] =  ct * a + st * b;
      W[sB * PDIM + col] = -st * a + ct * b;
    }
    __syncthreads();  // next gate shares qubits -> rows
  }
}

// ---------------- Phase 2: dense f32 WMMA matmul (transposed B) ------------
// C[m][n] = sum_k A[m][k] * B(k,n)  with  B(k,n) = Bp[n*PDIM + k]
//   (valid for matmul 1 because rho is symmetric; for matmul 2 because the
//    second operand is W^T and W is row-major)
// Each wave computes a 16x32 tile (2 accumulators sharing one A fragment).
// Block = 8 waves as 2(M) x 4(N) -> 32 x 128 tile.  All loads are b64 along
// the contiguous K direction; no guards in the hot loop.
template <int OUT_GUARD>
__global__ __launch_bounds__(256)
void wmma_matmul_tb_kernel(const float* __restrict__ A,
                           const float* __restrict__ Bp,
                           float* __restrict__ C,
                           int ldc) {
  const int lane  = threadIdx.x & 31;
  const int wave  = threadIdx.x >> 5;
  const int lmod  = lane & 15;
  const int lhalf = lane >> 4;
  const int m0 = (blockIdx.y * 2 + (wave & 1)) * 16;
  const int n0 = (blockIdx.x * 4 + (wave >> 1)) * 32;

  v8f acc0 = {};
  v8f acc1 = {};
  const float* arow  = A  + (m0 + lmod) * PDIM + 2 * lhalf;
  const float* brow0 = Bp + (n0 + lmod) * PDIM + 2 * lhalf;
  const float* brow1 = brow0 + 16 * PDIM;

#pragma unroll 4
  for (int k = 0; k < PDIM; k += 4) {
    // A fragment: lane l holds A[m0+l%16][k+2*(l/16) + {0,1}] -> one b64 load
    v2f a  = *(const v2f*)(arow + k);
    v2f b0 = *(const v2f*)(brow0 + k);
    v2f b1 = *(const v2f*)(brow1 + k);
    acc0 = __builtin_amdgcn_wmma_f32_16x16x4_f32(
        false, a, false, b0, (short)0, acc0, false, false);
    acc1 = __builtin_amdgcn_wmma_f32_16x16x4_f32(
        false, a, false, b1, (short)0, acc1, false, false);
  }

  // D/C layout: VGPR v, lane l -> (m = v + 8*(l/16), n = l%16)
#pragma unroll
  for (int v = 0; v < 8; ++v) {
    const int row = m0 + v + 8 * lhalf;
    const int c0  = n0 + lmod;
    const int c1  = c0 + 16;
    if (!OUT_GUARD || (row < NDIM && c0 < NDIM)) C[row * ldc + c0] = acc0[v];
    if (!OUT_GUARD || (row < NDIM && c1 < NDIM)) C[row * ldc + c1] = acc1[v];
  }
}

// ---------------------------------------------------------------------------
extern "C" void kernel_launch(void* const* d_in, const int* in_sizes, int n_in,
                              void* d_out, int out_size, void* d_ws, size_t ws_size,
                              hipStream_t stream) {
  const float* rho    = (const float*)d_in[0];  // 496*496 f32 (symmetric)
  const float* thetas = (const float*)d_in[1];  // 12 f32
  // d_in[2..4] = dense A/B/C stacks: unused (structure derived analytically)
  const int* u_idx = (const int*)d_in[5];       // 48 i32
  const int* p_idx = (const int*)d_in[6];       // 48 i32
  const int n_gates = in_sizes[5];

  float* W  = (float*)d_ws;                     // 512*512 f32 = 1 MB
  float* T  = W + PDIM * PDIM;                  // 512*512 f32 = 1 MB
  float* RP = T + PDIM * PDIM;                  // 512*512 f32 = 1 MB
  float* out = (float*)d_out;                   // 496*496 f32

  // Phase 0: rho -> zero-padded 512x512 (removes all guards from hot loops)
  pad_rho_kernel<<<PDIM * PDIM / 256, 256, 0, stream>>>(rho, RP);

  // Phase 1: W = U_48 ... U_1 (sparse Givens), one workgroup, L2-resident
  build_w_kernel<<<1, 1024, 0, stream>>>(thetas, u_idx, p_idx, n_gates, W);

  // Phase 2: T = W * rho_pad (rho symmetric -> transposed access exact);
  //          out = T * W^T.  512x512x512 f32 WMMA each, fully L2-resident.
  dim3 grid(PDIM / 128, PDIM / 32);             // 4 x 16 workgroups
  wmma_matmul_tb_kernel<0><<<grid, 256, 0, stream>>>(W, RP, T, PDIM);
  wmma_matmul_tb_kernel<1><<<grid, 256, 0, stream>>>(T, W, out, NDIM);
}